// CellAnnotator_89369679495685
// MI455X (gfx1250) — compile-verified
//
#include <hip/hip_runtime.h>

#define H 768
#define W 768
#define C 64

typedef __attribute__((ext_vector_type(2))) float v2f;
typedef __attribute__((ext_vector_type(8))) float v8f;

// ---------------------------------------------------------------------------
// Kernel A: spatially-varying 8x8 local conv with sigmoid(x) weights.
// Tile: 4 rows x 8 cols of pixels per 256-thread block.
// ---------------------------------------------------------------------------
#define TH 4
#define TW 8
#define PR (TH + 7)   // 11 padded rows   (pt=3, pb=4)
#define PC (TW + 7)   // 15 padded cols   (pl=3, pr=4)

__global__ __launch_bounds__(256) void att_kernel(const float* __restrict__ x,
                                                  float* __restrict__ y) {
    __shared__ float sx[PR][PC][C];     // 42.2 KB padded halo
    __shared__ float sw[TH][TW][C];     //  8.0 KB sigmoid weights

    const int tid = threadIdx.x;
    const int j0  = blockIdx.x * TW;
    const int i0  = blockIdx.y * TH;

    // Stage padded halo (zero fill outside image). 10560 floats.
    for (int idx = tid; idx < PR * PC * C; idx += 256) {
        int a = idx / (PC * C);
        int r = idx - a * (PC * C);
        int b = r / C;
        int c = r - b * C;
        int gi = i0 + a - 3;
        int gj = j0 + b - 3;
        float v = 0.f;
        if (gi >= 0 && gi < H && gj >= 0 && gj < W)
            v = x[((size_t)gi * W + gj) * C + c];
        sx[a][b][c] = v;
    }
    __syncthreads();

    // Per-pixel weights = sigmoid(center values). 2048 floats.
    for (int idx = tid; idx < TH * TW * C; idx += 256) {
        int p = idx >> 6;
        int c = idx & 63;
        int pi = p / TW;
        int pj = p - pi * TW;
        float v = sx[pi + 3][pj + 3][c];
        sw[pi][pj][c] = 1.0f / (1.0f + __expf(-v));
    }
    __syncthreads();

    // Each thread: one channel, one tile row, all 8 cols.
    const int c  = tid & 63;
    const int pi = tid >> 6;     // 0..3
    for (int pj = 0; pj < TW; ++pj) {
        float acc = 0.f;
        #pragma unroll
        for (int di = 0; di < 8; ++di) {
            #pragma unroll
            for (int dj = 0; dj < 8; ++dj) {
                acc += sx[pi + di][pj + dj][c] * sw[pi][pj][di * 8 + dj];
            }
        }
        y[((size_t)(i0 + pi) * W + (j0 + pj)) * C + c] = acc;
    }
}

// ---------------------------------------------------------------------------
// Kernel B: 4x (x @ W + b -> relu -> layernorm) then x @ w_out + b_out.
// 8 waves/block, 16 pixels/wave, WMMA f32 16x16x4 (M=16 pix, N=64, K=64).
// Reads att result from d_out x0 region and overwrites it in place.
// ---------------------------------------------------------------------------
__global__ __launch_bounds__(256) void mlp_kernel(const float* __restrict__ Ws,
                                                  const float* __restrict__ bs,
                                                  const float* __restrict__ lns,
                                                  const float* __restrict__ lnb,
                                                  const float* __restrict__ wout,
                                                  const float* __restrict__ bout,
                                                  float* __restrict__ x0g,
                                                  float* __restrict__ outg) {
    __shared__ float sWt[64][66];        // W transposed, padded: 16.9 KB
    __shared__ float sAct[8][16][66];    // per-wave activation staging: 33.8 KB
    __shared__ float sB[4][64], sLs[4][64], sLb[4][64], sWout[64];

    const int tid  = threadIdx.x;
    const int wv   = tid >> 5;
    const int lane = tid & 31;
    const int pixBase = blockIdx.x * 128 + wv * 16;

    // Small params (one pass; tid covers 0..255 exactly).
    ((float*)sB)[tid]  = bs[tid];
    ((float*)sLs)[tid] = lns[tid];
    ((float*)sLb)[tid] = lnb[tid];
    if (tid < 64) sWout[tid] = wout[tid];

    // Stage this wave's 16 activation rows (coalesced float2).
    const float* src = x0g + (size_t)pixBase * C;
    for (int i = 0; i < 16; ++i) {
        int idx2 = lane + i * 32;            // 0..511 float2s
        int m  = idx2 >> 5;
        int ch = (idx2 & 31) << 1;
        float2 v = *(const float2*)&src[m * C + ch];
        sAct[wv][m][ch]     = v.x;
        sAct[wv][m][ch + 1] = v.y;
    }

    const int m    = lane & 15;              // pixel row (A) / out column (B)
    const int kofs = (lane >> 4) << 1;       // K sub-offset per ISA f32 A layout
    const int mrow = m + ((lane >> 4) << 3); // unused; D-layout pixel row is r + 8*(lane>>4)

    for (int l = 0; l < 4; ++l) {
        __syncthreads();                     // protect sWt + staging
        const float* Wl = Ws + l * 4096;
        for (int idx = tid; idx < 4096; idx += 256) {
            int k = idx >> 6;
            int n = idx & 63;
            sWt[n][k] = Wl[idx];             // transpose: sWt[n][k] = W[k][n]
        }
        __syncthreads();

        // A fragments for all 16 K-steps (same across N tiles).
        v2f af[16];
        #pragma unroll
        for (int t = 0; t < 16; ++t) {
            int kk = t * 4 + kofs;
            af[t].x = sAct[wv][m][kk];
            af[t].y = sAct[wv][m][kk + 1];
        }

        v8f acc[4] = {};
        #pragma unroll
        for (int n = 0; n < 4; ++n) {
            const int nn = n * 16 + m;
            #pragma unroll
            for (int t = 0; t < 16; ++t) {
                int kk = t * 4 + kofs;
                v2f bf;
                bf.x = sWt[nn][kk];
                bf.y = sWt[nn][kk + 1];
                acc[n] = __builtin_amdgcn_wmma_f32_16x16x4_f32(
                    false, af[t], false, bf, (short)0, acc[n], false, false);
            }
        }

        // Bias + ReLU.
        #pragma unroll
        for (int n = 0; n < 4; ++n) {
            float bv = sB[l][n * 16 + m];
            #pragma unroll
            for (int r = 0; r < 8; ++r) {
                float t = acc[n][r] + bv;
                acc[n][r] = t > 0.f ? t : 0.f;
            }
        }

        // LayerNorm over 64 channels: 4 per-lane partials + 16-lane xor reduce.
        float mu[8], rs[8];
        #pragma unroll
        for (int r = 0; r < 8; ++r) {
            float s = acc[0][r] + acc[1][r] + acc[2][r] + acc[3][r];
            s += __shfl_xor(s, 1, 32);
            s += __shfl_xor(s, 2, 32);
            s += __shfl_xor(s, 4, 32);
            s += __shfl_xor(s, 8, 32);
            mu[r] = s * (1.0f / 64.0f);
            float d0 = acc[0][r] - mu[r];
            float d1 = acc[1][r] - mu[r];
            float d2 = acc[2][r] - mu[r];
            float d3 = acc[3][r] - mu[r];
            float v = d0 * d0 + d1 * d1 + d2 * d2 + d3 * d3;
            v += __shfl_xor(v, 1, 32);
            v += __shfl_xor(v, 2, 32);
            v += __shfl_xor(v, 4, 32);
            v += __shfl_xor(v, 8, 32);
            rs[r] = rsqrtf(v * (1.0f / 64.0f) + 1e-6f);
        }

        // Scale/bias, write normalized activations back to staging.
        #pragma unroll
        for (int n = 0; n < 4; ++n) {
            int chn = n * 16 + m;
            float sc = sLs[l][chn];
            float bi = sLb[l][chn];
            #pragma unroll
            for (int r = 0; r < 8; ++r) {
                sAct[wv][r + ((lane >> 4) << 3)][chn] =
                    (acc[n][r] - mu[r]) * rs[r] * sc + bi;
            }
        }
        (void)mrow;
    }
    __syncthreads();

    // Write final x0 (coalesced float2).
    float* dst = x0g + (size_t)pixBase * C;
    for (int i = 0; i < 16; ++i) {
        int idx2 = lane + i * 32;
        int mm = idx2 >> 5;
        int ch = (idx2 & 31) << 1;
        float2 v;
        v.x = sAct[wv][mm][ch];
        v.y = sAct[wv][mm][ch + 1];
        *(float2*)&dst[mm * C + ch] = v;
    }

    // Final projection: out = x0 @ w_out + b_out (one pixel per lane<16).
    if (lane < 16) {
        float s = 0.f;
        #pragma unroll
        for (int ch = 0; ch < 64; ++ch) s += sAct[wv][lane][ch] * sWout[ch];
        outg[pixBase + lane] = s + bout[0];
    }
}

// ---------------------------------------------------------------------------
extern "C" void kernel_launch(void* const* d_in, const int* in_sizes, int n_in,
                              void* d_out, int out_size, void* d_ws, size_t ws_size,
                              hipStream_t stream) {
    const float* x    = (const float*)d_in[0];
    const float* Ws   = (const float*)d_in[1];
    const float* bs   = (const float*)d_in[2];
    const float* lns  = (const float*)d_in[3];
    const float* lnb  = (const float*)d_in[4];
    const float* wout = (const float*)d_in[5];
    const float* bout = (const float*)d_in[6];

    float* x0g  = (float*)d_out;
    float* outg = x0g + (size_t)H * W * C;

    dim3 gA(W / TW, H / TH);
    att_kernel<<<gA, 256, 0, stream>>>(x, x0g);

    mlp_kernel<<<(H * W) / 128, 256, 0, stream>>>(Ws, bs, lns, lnb, wout, bout,
                                                  x0g, outg);
}